// AngularMarginLoss_82772609728800
// MI455X (gfx1250) — compile-verified
//
#include <hip/hip_runtime.h>
#include <hip/hip_bf16.h>
#include <math.h>
#include <stdint.h>

// ---------------------------------------------------------------------------
// AAM-softmax (ArcFace) loss on MI455X / gfx1250.
// 2*B*C*D = 105 GFLOP fp32 GEMM fused with exp/row-sum (logits never stored).
// fp32 tensor path: V_WMMA_F32_16X16X4_F32 (bf16 would corrupt exp(30*cos)).
// W (102.4MB) < L2 (192MB): 4-wave blocks share each staged W tile (64 emb
// rows) -> W re-read factor 32x from L2 (~3.3GB), HBM traffic ~105MB.
// W tiles double-buffered in LDS via GLOBAL_LOAD_ASYNC_TO_LDS_B128 (ASYNCcnt)
// so the tensor pipe overlaps the DMA; falls back to sync copies if the async
// builtins are not declared by this toolchain.
// ---------------------------------------------------------------------------

#define B_SAMP 2048
#define D_DIM  256
#define C_CLS  100000
#define SCALE_F 30.0f
#define MARGIN_F 0.2f
#define EPS_F 1e-6f

#define NT 25            // 16-wide column tiles per block
#define LDA 260          // padded LDS row stride (260 % 64 == 4 -> conflict-free)
#define WAVES 4          // waves per block, one 16-row m-subtile each
#define MROWS (16 * WAVES)

#if defined(__has_builtin)
#if __has_builtin(__builtin_amdgcn_global_load_async_to_lds_b128) && \
    __has_builtin(__builtin_amdgcn_s_wait_asynccnt)
#define HAS_ASYNC_LDS 1
#endif
#endif
#ifndef HAS_ASYNC_LDS
#define HAS_ASYNC_LDS 0
#endif

typedef __attribute__((ext_vector_type(2))) float v2f;
typedef __attribute__((ext_vector_type(8))) float v8f;

#if HAS_ASYNC_LDS
// Builtin signature (from hipcc diagnostic): param0 is a pointer to a 16-byte
// int vector in the global address space; param1 the LDS-side counterpart.
typedef int v4i_ __attribute__((vector_size(4 * sizeof(int))));
typedef __attribute__((address_space(1))) v4i_ as1_v4i;
typedef __attribute__((address_space(3))) v4i_ as3_v4i;

// One wave-wide async DMA: 32 lanes x 16B = 512B global -> LDS, ASYNCcnt++.
// LDS addrspace(3) pointer built from the generic address' low 32 bits
// (generic LDS addr[31:0] == LDS byte offset per the aperture mapping).
__device__ __forceinline__ void async_b128(const float* g, float* l) {
    __builtin_amdgcn_global_load_async_to_lds_b128(
        (as1_v4i*)(uintptr_t)g,
        (as3_v4i*)(uint32_t)(uintptr_t)l,
        0, 0);
}
#endif

// ---- Kernel 1: inverse row norms of W -------------------------------------
__global__ __launch_bounds__(256) void wnorm_kernel(const float* __restrict__ W,
                                                    float* __restrict__ rnorm) {
    const int lane = threadIdx.x & 31;
    const int wv   = threadIdx.x >> 5;
    const int row  = blockIdx.x * 8 + wv;
    if (row >= C_CLS) return;
    const float* p = W + (size_t)row * D_DIM;
    float ss = 0.0f;
#pragma unroll
    for (int i = 0; i < 8; ++i) {
        float v = p[lane + i * 32];
        ss += v * v;
    }
    for (int off = 16; off > 0; off >>= 1) ss += __shfl_xor(ss, off, 32);
    if (lane == 0) rnorm[row] = 1.0f / fmaxf(sqrtf(ss), 1e-12f);
}

// ---- Kernel 2: zero the per-sample exp-sum accumulators -------------------
__global__ void zero_kernel(float* __restrict__ S, int n) {
    int i = blockIdx.x * blockDim.x + threadIdx.x;
    if (i < n) S[i] = 0.0f;
}

// ---- Kernel 3: target logit -> numerator + exp(target) --------------------
__global__ __launch_bounds__(256) void target_kernel(const float* __restrict__ emb,
                                                     const float* __restrict__ W,
                                                     const int* __restrict__ tgt,
                                                     const float* __restrict__ rnorm,
                                                     float* __restrict__ num,
                                                     float* __restrict__ etg) {
    const int lane = threadIdx.x & 31;
    const int wv   = threadIdx.x >> 5;
    const int b    = blockIdx.x * 8 + wv;
    if (b >= B_SAMP) return;
    const int t = tgt[b];
    const float* e  = emb + (size_t)b * D_DIM;
    const float* wr = W   + (size_t)t * D_DIM;
    float d = 0.0f;
#pragma unroll
    for (int i = 0; i < 8; ++i) d += e[lane + i * 32] * wr[lane + i * 32];
    for (int off = 16; off > 0; off >>= 1) d += __shfl_xor(d, off, 32);
    if (lane == 0) {
        float tc = fminf(fmaxf(d * rnorm[t], -1.0f), 1.0f);
        num[b] = SCALE_F * cosf(acosf(tc) + MARGIN_F);
        etg[b] = expf(SCALE_F * tc);
    }
}

// ---- Kernel 4: fused GEMM(WMMA fp32) + exp + row-sum ----------------------
// grid = (B/64, C/(16*NT)), block = 128 (4 waves; wave w owns m-subtile w).
// No divergence in the WMMA region -> EXEC all-ones as the ISA requires.
__global__ __launch_bounds__(128) void gemm_expsum_kernel(const float* __restrict__ emb,
                                                          const float* __restrict__ W,
                                                          const float* __restrict__ rnorm,
                                                          float* __restrict__ S) {
    __shared__ float lds_a[MROWS * LDA];    // emb tile 64 x 256 (padded)
    __shared__ float lds_b[2][16 * LDA];    // W tile double buffer

    const int tid  = threadIdx.x;
    const int lane = tid & 31;
    const int wv   = tid >> 5;
    const int m0   = blockIdx.x * MROWS;
    const int c0   = blockIdx.y * (16 * NT);

#if HAS_ASYNC_LDS
    // Issue emb tile asyncs (64 rows contiguous): 32 b128 per wave.
    {
        const float* ga = emb + (size_t)m0 * D_DIM;
        for (int q = tid; q < MROWS * 64; q += 128) {
            int r = q >> 6, c4 = (q & 63) << 2;
            async_b128(ga + q * 4, &lds_a[r * LDA + c4]);
        }
    }
    // Issue W tile 0 asyncs: 8 b128 per wave.
    {
        const float* gb = W + (size_t)c0 * D_DIM;
        for (int q = tid; q < 1024; q += 128) {
            int r = q >> 6, c4 = (q & 63) << 2;
            async_b128(gb + q * 4, &lds_b[0][r * LDA + c4]);
        }
    }
#else
    // Fallback: synchronous emb tile copy.
    {
        const float4* ga = reinterpret_cast<const float4*>(emb + (size_t)m0 * D_DIM);
        for (int q = tid; q < MROWS * 64; q += 128) {
            int r = q >> 6, c4 = (q & 63) << 2;
            *reinterpret_cast<float4*>(&lds_a[r * LDA + c4]) = ga[q];
        }
    }
#endif

    const int m    = lane & 15;        // A row / B col within tile
    const int koff = (lane >> 4) << 1; // K sub-offset per WMMA fp32 A/B layout
    const float* pa = &lds_a[(wv * 16 + m) * LDA + koff];

    float rowacc[8];
#pragma unroll
    for (int j = 0; j < 8; ++j) rowacc[j] = 0.0f;

    for (int nt = 0; nt < NT; ++nt) {
        const int col0 = c0 + nt * 16;

#if HAS_ASYNC_LDS
        // Kick off next W tile into the other buffer, then retire current.
        if (nt + 1 < NT) {
            const float* gb = W + (size_t)(col0 + 16) * D_DIM;
            float* dbuf = lds_b[(nt + 1) & 1];
            for (int q = tid; q < 1024; q += 128) {
                int r = q >> 6, c4 = (q & 63) << 2;
                async_b128(gb + q * 4, &dbuf[r * LDA + c4]);
            }
            __builtin_amdgcn_s_wait_asynccnt(8);  // leave next tile in flight
        } else {
            __builtin_amdgcn_s_wait_asynccnt(0);
        }
        __syncthreads();                           // publish tile to all waves
        const float* pb = &lds_b[nt & 1][m * LDA + koff];
#else
        __syncthreads();                           // readers done before refill
        {
            const float4* gb = reinterpret_cast<const float4*>(W + (size_t)col0 * D_DIM);
            for (int q = tid; q < 1024; q += 128) {
                int r = q >> 6, c4 = (q & 63) << 2;
                *reinterpret_cast<float4*>(&lds_b[0][r * LDA + c4]) = gb[q];
            }
        }
        __syncthreads();
        const float* pb = &lds_b[0][m * LDA + koff];
#endif

        v8f acc = {0.f, 0.f, 0.f, 0.f, 0.f, 0.f, 0.f, 0.f};
#pragma unroll 8
        for (int k = 0; k < D_DIM; k += 4) {
            v2f a; a[0] = pa[k]; a[1] = pa[k + 1];
            v2f b; b[0] = pb[k]; b[1] = pb[k + 1];
            // D = A(16x4,f32) * B(4x16,f32) + C : fp32 tensor op
            acc = __builtin_amdgcn_wmma_f32_16x16x4_f32(
                false, a, false, b, (short)0, acc, false, false);
        }

        // Epilogue: cos = dot * rnorm[col], clip, exp(30x), row accumulate.
        const float rn = rnorm[col0 + m];
#pragma unroll
        for (int j = 0; j < 8; ++j) {
            float cosv = fminf(fmaxf(acc[j] * rn, -1.0f), 1.0f);
            rowacc[j] += expf(SCALE_F * cosv);
        }
#if HAS_ASYNC_LDS
        __syncthreads();   // all waves done reading before buffer is refilled
#endif
    }

    // C-tile layout: VGPR j = row j (lanes 0-15) / row j+8 (lanes 16-31),
    // N = lane&15. XOR-reduce within each 16-lane half.
#pragma unroll
    for (int j = 0; j < 8; ++j) {
        float v = rowacc[j];
        v += __shfl_xor(v, 1, 32);
        v += __shfl_xor(v, 2, 32);
        v += __shfl_xor(v, 4, 32);
        v += __shfl_xor(v, 8, 32);
        rowacc[j] = v;
    }
    if ((lane & 15) == 0) {
        const int half = (lane >> 4) * 8;
        const int base = m0 + wv * 16 + half;
#pragma unroll
        for (int j = 0; j < 8; ++j)
            atomicAdd(&S[base + j], rowacc[j]);
    }
}

// ---- Kernel 5: final loss reduction ---------------------------------------
__global__ __launch_bounds__(256) void loss_kernel(const float* __restrict__ S,
                                                   const float* __restrict__ num,
                                                   const float* __restrict__ etg,
                                                   float* __restrict__ out) {
    __shared__ float red[8];
    const int t = threadIdx.x;
    float p = 0.0f;
    for (int b = t; b < B_SAMP; b += 256) {
        float excl = S[b] - etg[b];               // masked sum over c != target
        float den  = expf(num[b]) + excl;
        p += num[b] - logf(den + EPS_F);
    }
    for (int off = 16; off > 0; off >>= 1) p += __shfl_xor(p, off, 32);
    if ((t & 31) == 0) red[t >> 5] = p;
    __syncthreads();
    if (t == 0) {
        float s = 0.0f;
#pragma unroll
        for (int i = 0; i < 8; ++i) s += red[i];
        out[0] = -s / (float)B_SAMP;
    }
}

// ---------------------------------------------------------------------------
extern "C" void kernel_launch(void* const* d_in, const int* in_sizes, int n_in,
                              void* d_out, int out_size, void* d_ws, size_t ws_size,
                              hipStream_t stream) {
    const float* emb = (const float*)d_in[0];   // [B, D] fp32
    const float* W   = (const float*)d_in[1];   // [C, D] fp32
    const int*   tgt = (const int*)d_in[2];     // [B]
    float* out = (float*)d_out;

    float* ws    = (float*)d_ws;
    float* rnorm = ws;                          // [C]
    float* S     = ws + C_CLS;                  // [B]
    float* num   = ws + C_CLS + B_SAMP;         // [B]
    float* etg   = ws + C_CLS + 2 * B_SAMP;     // [B]

    // 1) inverse row norms of W (8 rows per 256-thread block)
    wnorm_kernel<<<(C_CLS + 7) / 8, 256, 0, stream>>>(W, rnorm);
    // 2) zero exp-sum accumulators
    zero_kernel<<<(B_SAMP + 255) / 256, 256, 0, stream>>>(S, B_SAMP);
    // 3) target logits -> numerator + exp(tgt)
    target_kernel<<<B_SAMP / 8, 256, 0, stream>>>(emb, W, tgt, rnorm, num, etg);
    // 4) fused fp32 WMMA GEMM + exp + row sums (async double-buffered W tiles)
    dim3 grid(B_SAMP / MROWS, C_CLS / (16 * NT));
    gemm_expsum_kernel<<<grid, 128, 0, stream>>>(emb, W, rnorm, S);
    // 5) final scalar loss
    loss_kernel<<<1, 256, 0, stream>>>(S, num, etg, out);
}